// ResidualGATv2Receiver_20194936226249
// MI455X (gfx1250) — compile-verified
//
#include <hip/hip_runtime.h>
#include <hip/hip_bf16.h>
#include <stdint.h>

// ---------------------------------------------------------------------------
// ResidualGATv2 for MI455X (gfx1250, wave32).
// - Dense transforms: V_WMMA_F32_16X16X4_F32 (full fp32; GEMM work ~8 GFLOP,
//   negligible vs the L2-resident edge gather/scatter phase).
// - Weight staging: Tensor Data Mover (TENSOR_LOAD_TO_LDS) with LDS row
//   padding (stride 132 dwords) for conflict-free WMMA B-fragment reads.
// - Edge phase: wave-per-(edge,head), coalesced 128B gathers, 3-pass
//   scatter-softmax with float atomic-max (monotone int trick) + f32 atomics.
// ---------------------------------------------------------------------------

typedef __attribute__((ext_vector_type(2))) float v2f;
typedef __attribute__((ext_vector_type(8))) float v8f;
typedef __attribute__((ext_vector_type(4))) unsigned int tdm_u32x4;
typedef __attribute__((ext_vector_type(8))) int tdm_i32x8;
typedef __attribute__((ext_vector_type(4))) int tdm_i32x4;

#define HC    128
#define HEADS 4
#define CH    32
#define WPAD  132   // 128-dword W row + 4-dword TDM pad -> conflict-free LDS reads

// ---------------- fill ----------------
__global__ void fill_u32_kernel(unsigned* __restrict__ p, unsigned v, int n) {
  int i = blockIdx.x * blockDim.x + threadIdx.x;
  if (i < n) p[i] = v;
}

// ---------------- column stats (mean / unbiased std) ----------------
__global__ void col_stats_kernel(const float* __restrict__ x, float* __restrict__ stats,
                                 int N, int rowsPerBlock) {
  int c = threadIdx.x;                 // 128 threads = 128 columns
  int r0 = blockIdx.x * rowsPerBlock;
  int r1 = r0 + rowsPerBlock; if (r1 > N) r1 = N;
  float s = 0.f, s2 = 0.f;
  for (int r = r0; r < r1; ++r) {
    float v = x[(size_t)r * HC + c];
    s += v; s2 += v * v;
  }
  atomicAdd(&stats[c], s);
  atomicAdd(&stats[HC + c], s2);
}

__global__ void col_finalize_kernel(float* __restrict__ stats, int N) {
  int c = threadIdx.x;
  float mean = stats[c] / (float)N;
  float var = (stats[HC + c] - (float)N * mean * mean) / (float)(N - 1);
  var = fmaxf(var, 0.f);
  stats[c] = mean;
  stats[HC + c] = 1.f / (sqrtf(var) + 1e-6f);
}

__global__ void normalize_kernel(const float* __restrict__ x, const float* __restrict__ stats,
                                 float* __restrict__ xn, int total) {
  int i = blockIdx.x * blockDim.x + threadIdx.x;
  if (i >= total) return;
  int c = i & (HC - 1);
  xn[i] = (x[i] - stats[c]) * stats[HC + c];
}

// ---------------- WMMA GEMM: C[N,128] = A[N,128] @ W[128,128]^T + bias ------
// W row-major [out][in] is DMA'd into LDS by the Tensor Data Mover with a
// 4-dword pad per 128-dword row (row stride WPAD). B-fragment for WMMA is then
// a contiguous float2: b = { W[col][k], W[col][k+1] }.
__global__ __launch_bounds__(256) void gemm_x128_nt_kernel(
    const float* __restrict__ A, const float* __restrict__ W,
    const float* __restrict__ bias, float* __restrict__ C, int N) {
  __shared__ float Ws[HC * WPAD];      // 67.6 KB (gfx1250: up to 320 KB/WG)

  if (threadIdx.x < 32) {              // wave 0 issues the TDM descriptor
    unsigned long long gaddr = (unsigned long long)(uintptr_t)W;
    unsigned int lds_addr = (unsigned int)(uintptr_t)Ws;   // low 32b = LDS offset
    tdm_u32x4 g0;
    g0[0] = 1u;                                        // count=1 (valid D#)
    g0[1] = lds_addr;                                  // lds_addr[63:32]
    g0[2] = (unsigned int)(gaddr & 0xFFFFFFFFull);     // global_addr lo
    g0[3] = (unsigned int)((gaddr >> 32) & 0x1FFFFFFull) | (2u << 30); // hi | type=2
    tdm_i32x8 g1;
    // data_size=4B (2<<16) | pad_enable (1<<20) | pad_interval=128dw (6<<22)
    // | pad_amount=4dw (3<<25)
    g1[0] = (int)((2u << 16) | (1u << 20) | (6u << 22) | (3u << 25));
    g1[1] = (int)(128u << 16);   // tensor_dim0 = 128   (bits 79:48 -> lo16 here)
    g1[2] = (int)(128u << 16);   // tensor_dim1 = 128   (bits 111:80 -> lo16 here)
    g1[3] = (int)(128u << 16);   // tile_dim0   = 128   (bits 127:112)
    g1[4] = (int)(128u);         // tile_dim1   = 128, tile_dim2 = 0
    g1[5] = (int)(128u);         // tensor_dim0_stride = 128 (lo32)
    g1[6] = (int)(16384u << 16); // tensor_dim1_stride = 16384 (lo16 at 223:208)
    g1[7] = 0;                   // tensor_dim1_stride hi
    tdm_i32x4 gz = {0, 0, 0, 0}; // 2-D tensor: groups 2/3 unused
#if __clang_major__ >= 23
    tdm_i32x8 gz8 = {0, 0, 0, 0, 0, 0, 0, 0};
    __builtin_amdgcn_tensor_load_to_lds(g0, g1, gz, gz, gz8, 0);
#else
    __builtin_amdgcn_tensor_load_to_lds(g0, g1, gz, gz, 0);
#endif
    __builtin_amdgcn_s_wait_tensorcnt(0);
  }
  __syncthreads();                     // publish LDS tile to all 8 waves

  int tid = threadIdx.x;
  int wave = tid >> 5, lane = tid & 31;
  int m = lane & 15, g = lane >> 4;    // A row within tile; K-group select
  int nTiles = (N + 15) >> 4;
  int rowTile = blockIdx.x * 8 + wave; // one 16-row stripe per wave
  if (rowTile >= nTiles) return;       // wave-uniform: EXEC stays all-1 for WMMA
  int row0 = rowTile << 4;

  int arow = row0 + m; if (arow > N - 1) arow = N - 1;
  const float* Ap = A + (size_t)arow * HC;

  v8f zero = {0.f, 0.f, 0.f, 0.f, 0.f, 0.f, 0.f, 0.f};
  v8f acc[8];
#pragma unroll
  for (int t = 0; t < 8; ++t) acc[t] = zero;

#pragma unroll 4
  for (int kk = 0; kk < 32; ++kk) {
    int k = (kk << 2) + (g << 1);      // this lane covers K = k, k+1
    v2f a = *(const v2f*)&Ap[k];
#pragma unroll
    for (int t = 0; t < 8; ++t) {      // 8 column tiles of 16
      // B[k..k+1][n] = W[col][k..k+1], col = t*16 + (lane%16): contiguous pair
      v2f b = *(const v2f*)&Ws[((t << 4) + m) * WPAD + k];
      acc[t] = __builtin_amdgcn_wmma_f32_16x16x4_f32(
          false, a, false, b, (short)0, acc[t], false, false);
    }
  }

#pragma unroll
  for (int t = 0; t < 8; ++t) {
    float bv = bias[(t << 4) + m];
#pragma unroll
    for (int r = 0; r < 8; ++r) {
      int rr = row0 + r + (g << 3);    // C/D layout: M = r + 8*(lane/16)
      if (rr < N) C[(size_t)rr * HC + (t << 4) + m] = acc[t][r] + bv;
    }
  }
}

// ---------------- edge phase ----------------
__device__ __forceinline__ void atomicMaxFloat(float* addr, float val) {
  if (val >= 0.f) atomicMax((int*)addr, __float_as_int(val));
  else            atomicMin((unsigned int*)addr, __float_as_uint(val));
}

// pass 1: per-(edge,head) logit + per-dst running max
__global__ __launch_bounds__(256) void edge_logits_kernel(
    const long long* __restrict__ ei, const float* __restrict__ xl,
    const float* __restrict__ xr, const float* __restrict__ att,
    float* __restrict__ logits, float* __restrict__ nmax, int E, int EP) {
  int wid = (int)((blockIdx.x * (unsigned)blockDim.x + threadIdx.x) >> 5);
  int lane = threadIdx.x & 31;
  if (wid >= EP * HEADS) return;
  int e = wid >> 2, h = wid & 3;
  int src, dst;
  if (e < E) { src = (int)ei[e]; dst = (int)ei[E + e]; }
  else       { src = dst = e - E; }
  float v = xl[(size_t)src * HC + h * CH + lane] + xr[(size_t)dst * HC + h * CH + lane];
  v = v > 0.f ? v : 0.2f * v;                       // leaky_relu(0.2)
  float p = v * att[h * CH + lane];
#pragma unroll
  for (int off = 16; off; off >>= 1) p += __shfl_xor(p, off, 32);
  if (lane == 0) {
    logits[wid] = p;
    atomicMaxFloat(&nmax[dst * HEADS + h], p);
  }
}

// pass 2: ex = exp(logit - max); accumulate denom; store ex in place
__global__ void edge_expsum_kernel(const long long* __restrict__ ei,
                                   float* __restrict__ logits,
                                   const float* __restrict__ nmax,
                                   float* __restrict__ nsum, int E, int EP) {
  int i = blockIdx.x * blockDim.x + threadIdx.x;
  if (i >= EP * HEADS) return;
  int e = i >> 2, h = i & 3;
  int dst = (e < E) ? (int)ei[E + e] : e - E;
  float ex = __expf(logits[i] - nmax[dst * HEADS + h]);
  logits[i] = ex;
  atomicAdd(&nsum[dst * HEADS + h], ex);
}

// pass 3: agg[dst] += (ex/denom) * xl[src]
__global__ __launch_bounds__(256) void edge_aggregate_kernel(
    const long long* __restrict__ ei, const float* __restrict__ logits,
    const float* __restrict__ nsum, const float* __restrict__ xl,
    float* __restrict__ agg, int E, int EP) {
  int wid = (int)((blockIdx.x * (unsigned)blockDim.x + threadIdx.x) >> 5);
  int lane = threadIdx.x & 31;
  if (wid >= EP * HEADS) return;
  int e = wid >> 2, h = wid & 3;
  int src, dst;
  if (e < E) { src = (int)ei[e]; dst = (int)ei[E + e]; }
  else       { src = dst = e - E; }
  float alpha = logits[wid] / nsum[dst * HEADS + h];
  float val = alpha * xl[(size_t)src * HC + h * CH + lane];
  atomicAdd(&agg[(size_t)dst * HC + h * CH + lane], val);
}

// ---------------- epilogue: (+bias) -> BN(eval) -> ELU ----------------
__global__ void post_bn_elu_kernel(const float* __restrict__ agg,
                                   const float* __restrict__ bias,
                                   const float* __restrict__ g, const float* __restrict__ b,
                                   const float* __restrict__ m, const float* __restrict__ v,
                                   float* __restrict__ out, int total) {
  int i = blockIdx.x * blockDim.x + threadIdx.x;
  if (i >= total) return;
  int c = i & (HC - 1);
  float x = agg[i] + bias[c];
  x = (x - m[c]) * rsqrtf(v[c] + 1e-5f) * g[c] + b[c];
  out[i] = x > 0.f ? x : (__expf(x) - 1.f);         // ELU(alpha=1)
}

// ---------------- output head: out[n] = h[n] . linW + linb ----------------
__global__ __launch_bounds__(256) void head_kernel(const float* __restrict__ h,
                                                   const float* __restrict__ lw,
                                                   const float* __restrict__ lb,
                                                   float* __restrict__ out, int N) {
  int wid = (int)((blockIdx.x * (unsigned)blockDim.x + threadIdx.x) >> 5);
  int lane = threadIdx.x & 31;
  if (wid >= N) return;
  const float* hp = h + (size_t)wid * HC;
  float s = hp[lane]      * lw[lane]
          + hp[lane + 32] * lw[lane + 32]
          + hp[lane + 64] * lw[lane + 64]
          + hp[lane + 96] * lw[lane + 96];
#pragma unroll
  for (int off = 16; off; off >>= 1) s += __shfl_xor(s, off, 32);
  if (lane == 0) out[wid] = s + lb[0];
}

// ---------------------------------------------------------------------------
extern "C" void kernel_launch(void* const* d_in, const int* in_sizes, int n_in,
                              void* d_out, int out_size, void* d_ws, size_t ws_size,
                              hipStream_t stream) {
  const float*     x     = (const float*)d_in[0];
  const long long* ei    = (const long long*)d_in[1];
  const float* W1l  = (const float*)d_in[2];
  const float* b1l  = (const float*)d_in[3];
  const float* W1r  = (const float*)d_in[4];
  const float* b1r  = (const float*)d_in[5];
  const float* att1 = (const float*)d_in[6];
  const float* bias1= (const float*)d_in[7];
  const float* W2l  = (const float*)d_in[8];
  const float* b2l  = (const float*)d_in[9];
  const float* W2r  = (const float*)d_in[10];
  const float* b2r  = (const float*)d_in[11];
  const float* att2 = (const float*)d_in[12];
  const float* bias2= (const float*)d_in[13];
  const float* resW = (const float*)d_in[14];
  const float* resb = (const float*)d_in[15];
  const float* linW = (const float*)d_in[16];
  const float* linb = (const float*)d_in[17];
  const float* bn1g = (const float*)d_in[18];
  const float* bn1b = (const float*)d_in[19];
  const float* bn1m = (const float*)d_in[20];
  const float* bn1v = (const float*)d_in[21];
  const float* bn2g = (const float*)d_in[22];
  const float* bn2b = (const float*)d_in[23];
  const float* bn2m = (const float*)d_in[24];
  const float* bn2v = (const float*)d_in[25];

  const int N  = in_sizes[0] / HC;
  const int E  = in_sizes[1] / 2;
  const int EP = E + N;
  const size_t NHf = (size_t)N * HC;

  // workspace layout (floats)
  float* ws     = (float*)d_ws;
  float* xn     = ws;                       // N*128  layer input / layer output reuse
  float* xl     = xn + NHf;                 // N*128
  float* xr     = xl + NHf;                 // N*128
  float* agg    = xr + NHf;                 // N*128  (layer1: pre-seeded with residual)
  float* logits = agg + NHf;                // EP*4
  float* nmax   = logits + (size_t)EP * HEADS;  // N*4
  float* nsum   = nmax + (size_t)N * HEADS;     // N*4
  float* stats  = nsum + (size_t)N * HEADS;     // 256

  const int nTiles    = (N + 15) >> 4;
  const int gemmGrid  = (nTiles + 7) >> 3;
  const int totEH     = EP * HEADS;
  const int waveGrid  = (totEH + 7) >> 3;      // 8 waves / 256-thread block
  const int elemGridE = (totEH + 255) >> 8;
  const int elemGridN = ((int)NHf + 255) >> 8;
  const unsigned NEG_INF = 0xFF800000u;

  // ---- feature standardization ----
  fill_u32_kernel<<<1, 256, 0, stream>>>((unsigned*)stats, 0u, 2 * HC);
  {
    int rowsPer = 128;
    int blocks = (N + rowsPer - 1) / rowsPer;
    col_stats_kernel<<<blocks, HC, 0, stream>>>(x, stats, N, rowsPer);
  }
  col_finalize_kernel<<<1, HC, 0, stream>>>(stats, N);
  normalize_kernel<<<elemGridN, 256, 0, stream>>>(x, stats, xn, (int)NHf);

  // ---- layer 1 dense transforms (TDM + WMMA) ----
  gemm_x128_nt_kernel<<<gemmGrid, 256, 0, stream>>>(xn, W1l, b1l, xl, N);
  gemm_x128_nt_kernel<<<gemmGrid, 256, 0, stream>>>(xn, W1r, b1r, xr, N);
  gemm_x128_nt_kernel<<<gemmGrid, 256, 0, stream>>>(xn, resW, resb, agg, N); // seed agg = residual

  // ---- layer 1 edge softmax + aggregation ----
  fill_u32_kernel<<<(N * HEADS + 255) / 256, 256, 0, stream>>>((unsigned*)nmax, NEG_INF, N * HEADS);
  fill_u32_kernel<<<(N * HEADS + 255) / 256, 256, 0, stream>>>((unsigned*)nsum, 0u, N * HEADS);
  edge_logits_kernel<<<waveGrid, 256, 0, stream>>>(ei, xl, xr, att1, logits, nmax, E, EP);
  edge_expsum_kernel<<<elemGridE, 256, 0, stream>>>(ei, logits, nmax, nsum, E, EP);
  edge_aggregate_kernel<<<waveGrid, 256, 0, stream>>>(ei, logits, nsum, xl, agg, E, EP);
  post_bn_elu_kernel<<<elemGridN, 256, 0, stream>>>(agg, bias1, bn1g, bn1b, bn1m, bn1v, xn, (int)NHf);

  // ---- layer 2 dense transforms (TDM + WMMA) ----
  gemm_x128_nt_kernel<<<gemmGrid, 256, 0, stream>>>(xn, W2l, b2l, xl, N);
  gemm_x128_nt_kernel<<<gemmGrid, 256, 0, stream>>>(xn, W2r, b2r, xr, N);

  // ---- layer 2 edge softmax + aggregation ----
  fill_u32_kernel<<<elemGridN, 256, 0, stream>>>((unsigned*)agg, 0u, (int)NHf);
  fill_u32_kernel<<<(N * HEADS + 255) / 256, 256, 0, stream>>>((unsigned*)nmax, NEG_INF, N * HEADS);
  fill_u32_kernel<<<(N * HEADS + 255) / 256, 256, 0, stream>>>((unsigned*)nsum, 0u, N * HEADS);
  edge_logits_kernel<<<waveGrid, 256, 0, stream>>>(ei, xl, xr, att2, logits, nmax, E, EP);
  edge_expsum_kernel<<<elemGridE, 256, 0, stream>>>(ei, logits, nmax, nsum, E, EP);
  edge_aggregate_kernel<<<waveGrid, 256, 0, stream>>>(ei, logits, nsum, xl, agg, E, EP);
  post_bn_elu_kernel<<<elemGridN, 256, 0, stream>>>(agg, bias2, bn2g, bn2b, bn2m, bn2v, xn, (int)NHf);

  // ---- output head ----
  head_kernel<<<(N + 7) / 8, 256, 0, stream>>>(xn, linW, linb, (float*)d_out, N);
}